// SREL_rep_mu_3109556322489
// MI455X (gfx1250) — compile-verified
//
#include <hip/hip_runtime.h>
#include <math.h>

typedef float v2f __attribute__((ext_vector_type(2)));
typedef float v8f __attribute__((ext_vector_type(8)));

#define B_      64
#define LS_     1024
#define LW_     1024
#define LY_     2048
#define M_      16
#define D_      6144
#define H_      512
#define NSTEP   10
#define KSTATIC 4096          // wr|wi|y feature length
#define KDYN    2048          // sr|si feature length
#define BM      (B_ * M_)     // 1024 rows for the static GEMM

static __device__ __forceinline__ v8f wmma4(v2f a, v2f b, v8f c) {
  return __builtin_amdgcn_wmma_f32_16x16x4_f32(false, a, false, b,
                                               (short)0, c, false, false);
}

// ---------------------------------------------------------------------------
// Pack static features: A[(b*16+m), c] for c in [0,4096)
//   c <  1024 : w_M_re[b, c, m]
//   c <  2048 : w_M_im[b, c-1024, m]
//   else      : y_M[c-2048, m]
// ---------------------------------------------------------------------------
__global__ void srel_pack_static(const float* __restrict__ wre,
                                 const float* __restrict__ wim,
                                 const float* __restrict__ y,
                                 float* __restrict__ A) {
  int idx = blockIdx.x * blockDim.x + threadIdx.x;
  if (idx >= BM * KSTATIC) return;
  int c = idx & (KSTATIC - 1);
  int rowid = idx >> 12;            // KSTATIC == 4096 == 2^12
  int b = rowid >> 4, m = rowid & 15;
  float v;
  if (c < LW_)          v = wre[((size_t)b * LW_ + c) * M_ + m];
  else if (c < 2 * LW_) v = wim[((size_t)b * LW_ + (c - LW_)) * M_ + m];
  else                  v = y[(size_t)(c - 2 * LW_) * M_ + m];
  A[idx] = v;
}

// ---------------------------------------------------------------------------
// Pack a row-major weight matrix W[K x N] into WMMA B-fragment order:
//   Bpk[((k/4)*N + n)*4 + (k&3)] = W[k*N + n]
// so a lane's (k, k+1) pair is one aligned b64 load.
// ---------------------------------------------------------------------------
__global__ void srel_pack_B(const float* __restrict__ W, float* __restrict__ Bpk,
                            int K, int N) {
  int idx = blockIdx.x * blockDim.x + threadIdx.x;
  if (idx >= K * N) return;
  int k = idx / N, n = idx % N;
  Bpk[((size_t)(k >> 2) * N + n) * 4 + (k & 3)] = W[idx];
}

// ---------------------------------------------------------------------------
// f32 WMMA GEMM, packed B, 2x2 (32x32) output tiles per wave:
//   C[Mrows x N] = A[Mrows x K] * B[K x N] (+ bias)
// Per K-step: 2 A b64 loads + 2 packed-B b64 loads -> 4 independent v_wmma.
// ---------------------------------------------------------------------------
__global__ void srel_wmma_gemm2(const float* __restrict__ A,
                                const float* __restrict__ Bpk,
                                const float* __restrict__ bias,
                                float* __restrict__ C,
                                int Mrows, int N, int K) {
  int wave = (blockIdx.x * blockDim.x + threadIdx.x) >> 5;
  int lane = threadIdx.x & 31;
  int tilesN2 = N >> 5;
  int tilesM2 = Mrows >> 5;
  if (wave >= tilesM2 * tilesN2) return;   // wave-uniform exit
  int tm = wave / tilesN2, tn = wave % tilesN2;
  int r = lane & 15, half = lane >> 4;

  const float* Ap0 = A + (size_t)(tm * 32 + r) * K + half * 2;
  const float* Ap1 = Ap0 + (size_t)16 * K;
  int col0 = tn * 32 + r, col1 = col0 + 16;
  const float* Bp0 = Bpk + (size_t)col0 * 4 + half * 2;
  const float* Bp1 = Bpk + (size_t)col1 * 4 + half * 2;

  v8f acc00 = {0.f,0.f,0.f,0.f,0.f,0.f,0.f,0.f};
  v8f acc01 = acc00, acc10 = acc00, acc11 = acc00;
  for (int k = 0; k < K; k += 4) {
    v2f a0 = *(const v2f*)Ap0;
    v2f a1 = *(const v2f*)Ap1;
    v2f b0 = *(const v2f*)Bp0;
    v2f b1 = *(const v2f*)Bp1;
    acc00 = wmma4(a0, b0, acc00);
    acc01 = wmma4(a0, b1, acc01);
    acc10 = wmma4(a1, b0, acc10);
    acc11 = wmma4(a1, b1, acc11);
    Ap0 += 4; Ap1 += 4;
    Bp0 += (size_t)N * 4; Bp1 += (size_t)N * 4;
  }

  float bv0 = bias ? bias[col0] : 0.f;
  float bv1 = bias ? bias[col1] : 0.f;
  int row0 = tm * 32 + 8 * half;
#pragma unroll
  for (int j = 0; j < 8; ++j) {
    C[(size_t)(row0 + j) * N + col0]      = acc00[j] + bv0;
    C[(size_t)(row0 + j) * N + col1]      = acc01[j] + bv1;
    C[(size_t)(row0 + 16 + j) * N + col0] = acc10[j] + bv0;
    C[(size_t)(row0 + 16 + j) * N + col1] = acc11[j] + bv1;
  }
}

// ---------------------------------------------------------------------------
// eta second layer + phi update + s-emit, fully fused:
//   eta[b,l] = (V[b,:] @ eW2[:,l]) + csum[b]*eb2[l]
//   phi[b,l] -= eta ; sr=cos(phi), si=sin(phi)
//   out[b, t1, l, {0,1}] = {sr, si} ; dynA[b, {l, 1024+l}] = {sr, si}
// V is [B_ x H_]; W2pk is eW2 [H_ x LS_] in packed-B order.
// ---------------------------------------------------------------------------
__global__ void srel_eta_update_emit(const float* __restrict__ V,
                                     const float* __restrict__ W2pk,
                                     const float* __restrict__ eb2,
                                     const float* __restrict__ csum,
                                     float* __restrict__ phi,
                                     float* __restrict__ out,
                                     float* __restrict__ dynA, int t1) {
  const int N = LS_, K = H_;
  int wave = (blockIdx.x * blockDim.x + threadIdx.x) >> 5;
  int lane = threadIdx.x & 31;
  int tilesN2 = N >> 5;                 // 32
  if (wave >= (B_ >> 5) * tilesN2) return;
  int tm = wave / tilesN2, tn = wave % tilesN2;
  int r = lane & 15, half = lane >> 4;

  const float* Ap0 = V + (size_t)(tm * 32 + r) * K + half * 2;
  const float* Ap1 = Ap0 + (size_t)16 * K;
  int col0 = tn * 32 + r, col1 = col0 + 16;
  const float* Bp0 = W2pk + (size_t)col0 * 4 + half * 2;
  const float* Bp1 = W2pk + (size_t)col1 * 4 + half * 2;

  v8f acc00 = {0.f,0.f,0.f,0.f,0.f,0.f,0.f,0.f};
  v8f acc01 = acc00, acc10 = acc00, acc11 = acc00;
  for (int k = 0; k < K; k += 4) {
    v2f a0 = *(const v2f*)Ap0;
    v2f a1 = *(const v2f*)Ap1;
    v2f b0 = *(const v2f*)Bp0;
    v2f b1 = *(const v2f*)Bp1;
    acc00 = wmma4(a0, b0, acc00);
    acc01 = wmma4(a0, b1, acc01);
    acc10 = wmma4(a1, b0, acc10);
    acc11 = wmma4(a1, b1, acc11);
    Ap0 += 4; Ap1 += 4;
    Bp0 += (size_t)N * 4; Bp1 += (size_t)N * 4;
  }

  float e0 = eb2[col0], e1 = eb2[col1];
  int row0 = tm * 32 + 8 * half;
#pragma unroll
  for (int j = 0; j < 8; ++j) {
#pragma unroll
    for (int q = 0; q < 4; ++q) {
      int b   = row0 + (q & 1) * 16 + j;       // batch index
      int l   = (q & 2) ? col1 : col0;
      float eta = ((q == 0) ? acc00[j] : (q == 1) ? acc10[j]
                   : (q == 2) ? acc01[j] : acc11[j])
                  + csum[b] * ((q & 2) ? e1 : e0);
      float p = phi[(size_t)b * N + l] - eta;
      phi[(size_t)b * N + l] = p;
      float sr = cosf(p), si = sinf(p);
      size_t o = (((size_t)b * (NSTEP + 1) + t1) * LS_ + l) * 2;
      out[o] = sr;
      out[o + 1] = si;
      dynA[(size_t)b * KDYN + l] = sr;
      dynA[(size_t)b * KDYN + LS_ + l] = si;
    }
  }
}

// ---------------------------------------------------------------------------
// Step 0: phi = phi0 ; emit s(phi0) to out slot 0 and build dynA.
// ---------------------------------------------------------------------------
__global__ void srel_step0(const float* __restrict__ phi0,
                           float* __restrict__ phi,
                           float* __restrict__ out,
                           float* __restrict__ dynA) {
  int idx = blockIdx.x * blockDim.x + threadIdx.x;   // b*LS + l
  if (idx >= B_ * LS_) return;
  int b = idx >> 10, l = idx & (LS_ - 1);
  float p = phi0[idx];
  phi[idx] = p;
  float sr = cosf(p), si = sinf(p);                  // MODULUS == 1
  size_t o = (((size_t)b * (NSTEP + 1)) * LS_ + l) * 2;
  out[o] = sr;
  out[o + 1] = si;
  dynA[(size_t)b * KDYN + l] = sr;
  dynA[(size_t)b * KDYN + LS_ + l] = si;
}

// ---------------------------------------------------------------------------
// Fused per-step combine. One block per batch b, blockDim = H_ = 512.
// For each m: rho = <relu(dyn_r+st_r), rW2> + rb2 ; mu likewise ; c = mu*rho
//   V[b,h]  = sum_m c * relu(dyn_e + st_e) ;  csum[b] = sum_m c
// ---------------------------------------------------------------------------
__global__ void srel_fuse_reduce(const float* __restrict__ dyn_e,
                                 const float* __restrict__ dyn_r,
                                 const float* __restrict__ dyn_m,
                                 const float* __restrict__ st_e,
                                 const float* __restrict__ st_r,
                                 const float* __restrict__ st_m,
                                 const float* __restrict__ rW2,
                                 const float* __restrict__ rb2,
                                 const float* __restrict__ mW2,
                                 const float* __restrict__ mb2,
                                 float* __restrict__ V,
                                 float* __restrict__ csum) {
  __shared__ float red[H_];
  __shared__ float bc[2];
  int b = blockIdx.x;
  int h = threadIdx.x;
  float de = dyn_e[b * H_ + h];
  float dr = dyn_r[b * H_ + h];
  float dm = dyn_m[b * H_ + h];
  float wr2 = rW2[h], wm2 = mW2[h];
  float rb = rb2[0], mb = mb2[0];
  float vacc = 0.f, cacc = 0.f;

  for (int m = 0; m < M_; ++m) {
    size_t srow = ((size_t)(b * M_ + m)) * H_ + h;
    float hr = dr + st_r[srow]; hr = hr > 0.f ? hr : 0.f;
    float hm = dm + st_m[srow]; hm = hm > 0.f ? hm : 0.f;

    red[h] = hr * wr2;
    __syncthreads();
    for (int s = H_ / 2; s > 0; s >>= 1) {
      if (h < s) red[h] += red[h + s];
      __syncthreads();
    }
    if (h == 0) bc[0] = red[0];
    red[h] = hm * wm2;
    __syncthreads();
    for (int s = H_ / 2; s > 0; s >>= 1) {
      if (h < s) red[h] += red[h + s];
      __syncthreads();
    }
    if (h == 0) bc[1] = red[0];
    __syncthreads();

    float c = (bc[1] + mb) * (bc[0] + rb);
    float he = de + st_e[srow]; he = he > 0.f ? he : 0.f;
    vacc += c * he;
    cacc += c;
    __syncthreads();
  }
  V[b * H_ + h] = vacc;
  if (h == 0) csum[b] = cacc;
}

__global__ void srel_zero_f32(float* __restrict__ d, int n) {
  int i = blockIdx.x * blockDim.x + threadIdx.x;
  if (i < n) d[i] = 0.f;
}

// ---------------------------------------------------------------------------
extern "C" void kernel_launch(void* const* d_in, const int* in_sizes, int n_in,
                              void* d_out, int out_size, void* d_ws, size_t ws_size,
                              hipStream_t stream) {
  (void)in_sizes; (void)n_in; (void)out_size; (void)ws_size;
  const float* phi0 = (const float*)d_in[0];
  const float* wre  = (const float*)d_in[1];
  const float* wim  = (const float*)d_in[2];
  const float* y    = (const float*)d_in[3];
  const float* eW1  = (const float*)d_in[4];
  const float* eb1  = (const float*)d_in[5];
  const float* eW2  = (const float*)d_in[6];
  const float* eb2  = (const float*)d_in[7];
  const float* rW1  = (const float*)d_in[8];
  const float* rb1  = (const float*)d_in[9];
  const float* rW2  = (const float*)d_in[10];
  const float* rb2  = (const float*)d_in[11];
  const float* mW1  = (const float*)d_in[12];
  const float* mb1  = (const float*)d_in[13];
  const float* mW2  = (const float*)d_in[14];
  const float* mb2  = (const float*)d_in[15];
  float* out = (float*)d_out;

  // workspace carve-up (floats)
  float* ws     = (float*)d_ws;
  float* Astat  = ws;                               // 1024*4096
  float* Bpk_e  = Astat + (size_t)BM * KSTATIC;     // 6144*512 each (packed W1)
  float* Bpk_r  = Bpk_e + (size_t)D_ * H_;
  float* Bpk_m  = Bpk_r + (size_t)D_ * H_;
  float* Bpk_w2 = Bpk_m + (size_t)D_ * H_;          // 512*1024 (packed eW2)
  float* st_e   = Bpk_w2 + (size_t)H_ * LS_;        // 1024*512 each
  float* st_r   = st_e + (size_t)BM * H_;
  float* st_m   = st_r + (size_t)BM * H_;
  float* phi    = st_m + (size_t)BM * H_;           // 64*1024
  float* dynA   = phi + B_ * LS_;                   // 64*2048
  float* dh_e   = dynA + B_ * KDYN;                 // 64*512 each
  float* dh_r   = dh_e + B_ * H_;
  float* dh_m   = dh_r + B_ * H_;
  float* V      = dh_m + B_ * H_;                   // 64*512
  float* csum   = V + B_ * H_;                      // 64

  // 1) one-time packs: static features + weight matrices into fragment order
  srel_pack_static<<<(BM * KSTATIC + 255) / 256, 256, 0, stream>>>(wre, wim, y, Astat);
  srel_pack_B<<<((int)((size_t)D_ * H_) + 255) / 256, 256, 0, stream>>>(eW1, Bpk_e, D_, H_);
  srel_pack_B<<<((int)((size_t)D_ * H_) + 255) / 256, 256, 0, stream>>>(rW1, Bpk_r, D_, H_);
  srel_pack_B<<<((int)((size_t)D_ * H_) + 255) / 256, 256, 0, stream>>>(mW1, Bpk_m, D_, H_);
  srel_pack_B<<<((int)((size_t)H_ * LS_) + 255) / 256, 256, 0, stream>>>(eW2, Bpk_w2, H_, LS_);

  // 2) one-time static-half GEMMs (bias b1 folded here).
  //    Packed-B offset for W1 rows [2048, 6144): (2048/4)*H_*4 == 2048*H_.
  int wavesStat = (BM / 32) * (H_ / 32);            // 512 waves
  srel_wmma_gemm2<<<(wavesStat * 32 + 255) / 256, 256, 0, stream>>>(
      Astat, Bpk_e + (size_t)2048 * H_, eb1, st_e, BM, H_, KSTATIC);
  srel_wmma_gemm2<<<(wavesStat * 32 + 255) / 256, 256, 0, stream>>>(
      Astat, Bpk_r + (size_t)2048 * H_, rb1, st_r, BM, H_, KSTATIC);
  srel_wmma_gemm2<<<(wavesStat * 32 + 255) / 256, 256, 0, stream>>>(
      Astat, Bpk_m + (size_t)2048 * H_, mb1, st_m, BM, H_, KSTATIC);

  // 3) step 0 init: phi <- phi0, emit slot 0, build dynA
  srel_step0<<<(B_ * LS_ + 255) / 256, 256, 0, stream>>>(phi0, phi, out, dynA);

  int wavesDyn = (B_ / 32) * (H_ / 32);             // 32 waves
  int wavesEta = (B_ / 32) * (LS_ / 32);            // 64 waves
  for (int t = 0; t < NSTEP; ++t) {
    srel_wmma_gemm2<<<(wavesDyn * 32 + 255) / 256, 256, 0, stream>>>(
        dynA, Bpk_e, nullptr, dh_e, B_, H_, KDYN);
    srel_wmma_gemm2<<<(wavesDyn * 32 + 255) / 256, 256, 0, stream>>>(
        dynA, Bpk_r, nullptr, dh_r, B_, H_, KDYN);
    srel_wmma_gemm2<<<(wavesDyn * 32 + 255) / 256, 256, 0, stream>>>(
        dynA, Bpk_m, nullptr, dh_m, B_, H_, KDYN);
    srel_fuse_reduce<<<B_, H_, 0, stream>>>(dh_e, dh_r, dh_m, st_e, st_r, st_m,
                                            rW2, rb2, mW2, mb2, V, csum);
    // updates phi, emits slot t+1, rebuilds dynA for the next step
    srel_eta_update_emit<<<(wavesEta * 32 + 255) / 256, 256, 0, stream>>>(
        V, Bpk_w2, eb2, csum, phi, out, dynA, t + 1);
  }

  // zero mu_stack tail of the output
  srel_zero_f32<<<(B_ * NSTEP * M_ + 255) / 256, 256, 0, stream>>>(
      out + (size_t)B_ * (NSTEP + 1) * LS_ * 2, B_ * NSTEP * M_);
}